// ARAPOptimise_26164940767233
// MI455X (gfx1250) — compile-verified
//
#include <hip/hip_runtime.h>

// Problem constants from the reference
#define BB 8
#define NN 200000
#define KK 16
#define BN (BB * NN)
#define NGROUPS ((BN + 4) / 5)   // 5 vertices per wave-group (block-diagonal WMMA packing)

typedef float v2f __attribute__((ext_vector_type(2)));
typedef float v8f __attribute__((ext_vector_type(8)));

// ARAP energy, per vertex i:
//   E_i = sum_k w (|d0|^2 + |dp|^2) - 2 * tr(R_i S_i)
//   tr(R_i S_i) = s1 + s2 + sign(det S_i) * s3   (singular values of S_i)
// S_i (3x3) computed on the tensor core: 5 vertices packed block-diagonally into
// one 16x16 tile, contraction over 16 neighbors = 4 chained V_WMMA_F32_16X16X4_F32.
// Padding lanes (15/31) only write row/col 15 of the tile, which is never read,
// so the inner loop needs no exec masking at all.
__global__ __launch_bounds__(256) void arap_energy_kernel(
    const float* __restrict__ vr,   // verts_rest [B,N,3]
    const float* __restrict__ vd,   // verts_def  [B,N,3]
    const float* __restrict__ wgt,  // weights    [N,K]
    const int*   __restrict__ nbr,  // nbr_idx    [N,K]
    float* __restrict__ out)
{
    __shared__ float ldsC[8][16][17];   // per-wave 16x16 C spill (+1 pad vs bank conflicts)
    __shared__ float redBuf[8];

    const int lane  = threadIdx.x & 31;
    const int wid   = threadIdx.x >> 5;
    const int lhalf = lane & 15;
    const int v     = lhalf / 3;          // 0..5 (5 => padding lane 15/31)
    const int aComp = lhalf - 3 * v;      // xyz component this lane owns
    const int kOff  = (lane >= 16) ? 2 : 0;   // f32 WMMA A/B layout: hi lanes hold K+2
    const float m   = (v < 5) ? 1.0f : 0.0f;  // padding-lane mask for the scalar term

    const int wavesPerBlock = blockDim.x >> 5;
    int g = blockIdx.x * wavesPerBlock + wid;
    const int gStride = gridDim.x * wavesPerBlock;

    float acc = 0.0f;

    for (; g < NGROUPS; g += gStride) {
        // Flat (b*N+n) vertex id; clamp padding lane into bounds (its data is never read)
        int idx = g * 5 + v;
        idx = (idx > BN - 1) ? (BN - 1) : idx;
        const int b = idx / NN;
        const int n = idx - b * NN;

        const float* __restrict__ vrb  = vr + b * (NN * 3);   // this batch's rest verts
        const float* __restrict__ vdb  = vd + b * (NN * 3);   // this batch's def verts
        const int*   __restrict__ nrow = nbr + n * KK;        // adjacency row
        const float* __restrict__ wrow = wgt + n * KK;        // weight row

        const float cr = vrb[n * 3 + aComp];   // center rest (comp a)
        const float cd = vdb[n * 3 + aComp];   // center def  (comp a)

        // Streaming prefetch of the next group's adjacency/weight rows
        if ((n + 5) < NN) {
            __builtin_prefetch(nrow + 5 * KK, 0, 1);
            __builtin_prefetch(wrow + 5 * KK, 0, 1);
        }

        v8f C = {};   // 16x16 f32 accumulator: 5 block-diagonal 3x3 covariances

        #pragma unroll
        for (int t = 0; t < 4; ++t) {
            const int   k0 = 4 * t + kOff;
            const int   j0 = nrow[k0];
            const int   j1 = nrow[k0 + 1];
            const float w0 = wrow[k0];
            const float w1 = wrow[k0 + 1];
            const float r0 = vrb[j0 * 3 + aComp];
            const float r1 = vrb[j1 * 3 + aComp];
            const float f0 = vdb[j0 * 3 + aComp];
            const float f1 = vdb[j1 * 3 + aComp];
            const float d00 = cr - r0, d01 = cr - r1;   // rest edge comp
            const float e0  = cd - f0, e1  = cd - f1;   // def edge comp

            v2f A  = {d00, d01};               // d0 component a, neighbors k0,k0+1
            v2f Bm = {w0 * e0, w1 * e1};       // w*dp component a

            // constant part of energy: sum_k w (|d0|^2 + |dp|^2), masked for padding lanes
            acc += m * (w0 * (d00 * d00 + e0 * e0) + w1 * (d01 * d01 + e1 * e1));

            // D = A(16x4) * B(4x16) + C  -> 5 diagonal 3x3 blocks are the S_i
            C = __builtin_amdgcn_wmma_f32_16x16x4_f32(
                    /*neg_a=*/false, A, /*neg_b=*/false, Bm,
                    /*c_mod=*/(short)0, C, /*reuse_a=*/false, /*reuse_b=*/false);
        }

        // Spill C tile to this wave's LDS slice. C layout: vgpr r -> row (r or r+8), col = lane&15
        {
            const int rowO = (lane >> 4) << 3;   // 0 or 8
            #pragma unroll
            for (int r = 0; r < 8; ++r)
                ldsC[wid][rowO + r][lhalf] = C[r];
        }
        asm volatile("s_wait_dscnt 0x0" ::: "memory");   // wave-private LDS RAW

        // 5 lanes: closed-form singular values of their 3x3 S, accumulate -2*tr(R S)
        if (lane < 5) {
            const int d = 3 * lane;
            const float s00 = ldsC[wid][d+0][d+0], s01 = ldsC[wid][d+0][d+1], s02 = ldsC[wid][d+0][d+2];
            const float s10 = ldsC[wid][d+1][d+0], s11 = ldsC[wid][d+1][d+1], s12 = ldsC[wid][d+1][d+2];
            const float s20 = ldsC[wid][d+2][d+0], s21 = ldsC[wid][d+2][d+1], s22 = ldsC[wid][d+2][d+2];

            const float det = s00 * (s11 * s22 - s12 * s21)
                            - s01 * (s10 * s22 - s12 * s20)
                            + s02 * (s10 * s21 - s11 * s20);

            // M = S^T S (symmetric PSD); eigenvalues are squared singular values
            const float m00 = s00*s00 + s10*s10 + s20*s20;
            const float m11 = s01*s01 + s11*s11 + s21*s21;
            const float m22 = s02*s02 + s12*s12 + s22*s22;
            const float m01 = s00*s01 + s10*s11 + s20*s21;
            const float m02 = s00*s02 + s10*s12 + s20*s22;
            const float m12 = s01*s02 + s11*s12 + s21*s22;

            const float q  = (m00 + m11 + m22) * (1.0f / 3.0f);
            const float p1 = m01*m01 + m02*m02 + m12*m12;
            const float a0 = m00 - q, a1 = m11 - q, a2 = m22 - q;
            const float p2 = a0*a0 + a1*a1 + a2*a2 + 2.0f * p1;

            // Branchless trigonometric eigensolve; p2 clamped so p->0 gives l1=l2=l3=q
            const float p  = __builtin_sqrtf(fmaxf(p2, 1e-30f) * (1.0f / 6.0f));
            const float ip = 1.0f / p;
            const float b00 = a0 * ip, b11 = a1 * ip, b22 = a2 * ip;
            const float b01 = m01 * ip, b02 = m02 * ip, b12 = m12 * ip;
            float r = 0.5f * (b00 * (b11 * b22 - b12 * b12)
                            - b01 * (b01 * b22 - b12 * b02)
                            + b02 * (b01 * b12 - b11 * b02));
            r = fminf(fmaxf(r, -1.0f), 1.0f);
            const float phi = acosf(r) * (1.0f / 3.0f);
            const float l1 = q + 2.0f * p * __builtin_cosf(phi);
            const float l3 = q + 2.0f * p * __builtin_cosf(phi + 2.09439510239319549f); // +2pi/3
            const float l2 = 3.0f * q - l1 - l3;

            const float sg1 = __builtin_sqrtf(fmaxf(l1, 0.0f));
            const float sg2 = __builtin_sqrtf(fmaxf(l2, 0.0f));
            const float sg3 = __builtin_sqrtf(fmaxf(l3, 0.0f));  // smallest -> det-corrected
            const float trRS = sg1 + sg2 + ((det < 0.0f) ? -sg3 : sg3);
            acc -= 2.0f * trRS;
        }
    }

    // Block reduction: wave butterfly, then cross-wave via LDS, one atomic per block
    float tot = acc;
    #pragma unroll
    for (int off = 16; off > 0; off >>= 1)
        tot += __shfl_xor(tot, off, 32);
    if (lane == 0) redBuf[wid] = tot;
    __syncthreads();
    if (threadIdx.x == 0) {
        float s = 0.0f;
        for (int i = 0; i < wavesPerBlock; ++i) s += redBuf[i];
        atomicAdd(out, s);
    }
}

extern "C" void kernel_launch(void* const* d_in, const int* in_sizes, int n_in,
                              void* d_out, int out_size, void* d_ws, size_t ws_size,
                              hipStream_t stream) {
    (void)in_sizes; (void)n_in; (void)d_ws; (void)ws_size;
    const float* verts_rest = (const float*)d_in[0];
    const float* verts_def  = (const float*)d_in[1];
    const float* weights    = (const float*)d_in[2];
    const int*   nbr_idx    = (const int*)d_in[3];
    float* out = (float*)d_out;

    hipMemsetAsync(out, 0, (size_t)out_size * sizeof(float), stream);

    // 1024 blocks * 8 waves = 8192 waves; 320k groups -> ~39 groups/wave (grid-stride)
    arap_energy_kernel<<<dim3(1024), dim3(256), 0, stream>>>(
        verts_rest, verts_def, weights, nbr_idx, out);
}